// FUB_26310969655406
// MI455X (gfx1250) — compile-verified
//
#include <hip/hip_runtime.h>

typedef float v2f __attribute__((ext_vector_type(2)));
typedef float v8f __attribute__((ext_vector_type(8)));

#define N_IMG 5
#define NCH   256
#define HW    4096
#define PIX   (NCH * HW)   // 1,048,576 pixels per image

// ---------------------------------------------------------------------------
// Kernel 1: Wsum[c*256+d] = W1[c,d] + W2[c,d]
// (W2 itself needs no copy: conv_w+65536 already has the w[c*256+d] layout.)
// ---------------------------------------------------------------------------
__global__ __launch_bounds__(256) void prep_wsum(const float* __restrict__ w,
                                                 float* __restrict__ wsum) {
    const int i = blockIdx.x * blockDim.x + threadIdx.x;   // 0 .. 65535
    wsum[i] = w[i] + w[i + 65536];
}

// ---------------------------------------------------------------------------
// Kernel 2: fp32 GEMM via V_WMMA_F32_16X16X4_F32 with 2x4 register blocking.
//   out[d, hw] = sum_c w[c*256 + d] * xin[c*4096 + hw]
// Each wave computes a 32(d) x 64(hw) block = 2x4 WMMA tiles.
// Per K-step (K=4): 4 A-loads + 8 B-loads -> 8 WMMAs (1.5 loads/WMMA).
// grid.z in [0,10): z/5 selects matrix (0: A with W2, 1: Bf with Wsum), z%5 = image.
// Fragment layouts per CDNA5 ISA 7.12.2:
//   A 16x4 : lanes 0-15 -> M rows, v0/v1 = K0/K1; lanes 16-31 -> K2/K3
//   B 4x16 : lanes 0-15 -> N cols, v0/v1 = K0/K1; lanes 16-31 -> K2/K3
//   D 16x16: VGPR r, lane L -> M = r + 8*(L/16), N = L%16
// ---------------------------------------------------------------------------
__global__ __launch_bounds__(256) void gemm_wmma(const float* __restrict__ x,
                                                 const float* __restrict__ w2,
                                                 const float* __restrict__ wsum,
                                                 float* __restrict__ Abuf,
                                                 float* __restrict__ Bbuf) {
    const int z   = blockIdx.z;
    const int m   = z / N_IMG;               // 0 -> A, 1 -> Bf
    const int img = z % N_IMG;

    const float* __restrict__ w   = (m == 0) ? w2 : wsum;
    const float* __restrict__ xin = x + img * PIX;
    float* __restrict__ out       = ((m == 0) ? Abuf : Bbuf) + img * PIX;

    const int wave = threadIdx.x >> 5;
    const int lane = threadIdx.x & 31;
    const int half = lane >> 4;              // 0 or 1
    const int l16  = lane & 15;

    const int t   = blockIdx.x * 8 + wave;   // 0 .. 511 wave-blocks
    const int d0  = (t >> 6) << 5;           // 8 d-blocks of 32
    const int hw0 = (t & 63) << 6;           // 64 hw-blocks of 64

    v8f acc[2][4];
#pragma unroll
    for (int i = 0; i < 2; ++i)
#pragma unroll
        for (int j = 0; j < 4; ++j)
            acc[i][j] = (v8f){};

#pragma unroll 2
    for (int k = 0; k < NCH; k += 4) {
        const int c0 = k + 2 * half;

        v2f a[2], b[4];
#pragma unroll
        for (int i = 0; i < 2; ++i) {
            a[i].x = w[c0 * 256 + d0 + 16 * i + l16];        // A[M][K=c0]
            a[i].y = w[(c0 + 1) * 256 + d0 + 16 * i + l16];  // A[M][K=c0+1]
        }
#pragma unroll
        for (int j = 0; j < 4; ++j) {
            b[j].x = xin[c0 * HW + hw0 + 16 * j + l16];      // B[K=c0][N]
            b[j].y = xin[(c0 + 1) * HW + hw0 + 16 * j + l16];
        }
#pragma unroll
        for (int i = 0; i < 2; ++i)
#pragma unroll
            for (int j = 0; j < 4; ++j)
                acc[i][j] = __builtin_amdgcn_wmma_f32_16x16x4_f32(
                    /*neg_a=*/false, a[i], /*neg_b=*/false, b[j],
                    /*c_mod=*/(short)0, acc[i][j],
                    /*reuse_a=*/false, /*reuse_b=*/false);
    }

#pragma unroll
    for (int i = 0; i < 2; ++i)
#pragma unroll
        for (int j = 0; j < 4; ++j)
#pragma unroll
            for (int r = 0; r < 8; ++r)
                out[(d0 + 16 * i + r + 8 * half) * HW + hw0 + 16 * j + l16] =
                    acc[i][j][r];
}

// ---------------------------------------------------------------------------
// Kernel 3: per-pixel 5x5 edge matrix, row norm, threshold, weighted sum.
//   E[i,j] = |x[j,p] - A[i,p] - (Bf[j,p] + b[d])|
//   out[i,p] = sum_j (E[i,j] > 0.35 * max(||E_i||, 1e-12) ? E[i,j] : 0) * x[j,p]
// ---------------------------------------------------------------------------
__global__ __launch_bounds__(256) void fuse(const float* __restrict__ x,
                                            const float* __restrict__ conv_b,
                                            const float* __restrict__ Abuf,
                                            const float* __restrict__ Bbuf,
                                            float* __restrict__ out) {
    const int p = blockIdx.x * blockDim.x + threadIdx.x;   // 0 .. PIX-1
    const int d = (p >> 12) & 255;                         // channel of this pixel
    const float bd = conv_b[d];

    float xv[N_IMG], bf[N_IMG];
#pragma unroll
    for (int j = 0; j < N_IMG; ++j) {
        xv[j] = x[j * PIX + p];
        bf[j] = Bbuf[j * PIX + p] + bd;    // fold bias into Bf term
    }

#pragma unroll
    for (int i = 0; i < N_IMG; ++i) {
        const float ai = Abuf[i * PIX + p];
        float e[N_IMG];
        float s = 0.0f;
#pragma unroll
        for (int j = 0; j < N_IMG; ++j) {
            e[j] = fabsf(xv[j] - ai - bf[j]);
            s += e[j] * e[j];
        }
        const float thr = 0.35f * fmaxf(sqrtf(s), 1e-12f);
        float h = 0.0f;
#pragma unroll
        for (int j = 0; j < N_IMG; ++j) {
            h += (e[j] > thr) ? e[j] * xv[j] : 0.0f;
        }
        out[i * PIX + p] = h;
    }
}

// ---------------------------------------------------------------------------
// Launch. d_in order: x (5,1,256,64,64) f32; conv_w (512,256) f32; conv_b (256,) f32.
// d_out: (5,1,256,64,64) f32.  Workspace: Wsum (64K f) | A (5M f) | Bf (5M f).
// ---------------------------------------------------------------------------
extern "C" void kernel_launch(void* const* d_in, const int* in_sizes, int n_in,
                              void* d_out, int out_size, void* d_ws, size_t ws_size,
                              hipStream_t stream) {
    const float* x      = (const float*)d_in[0];
    const float* conv_w = (const float*)d_in[1];
    const float* conv_b = (const float*)d_in[2];
    float* out = (float*)d_out;

    float* ws   = (float*)d_ws;
    float* wsum = ws;                       // 65536 floats
    float* Abuf = ws + 65536;               // 5 * PIX floats
    float* Bbuf = Abuf + N_IMG * PIX;       // 5 * PIX floats

    prep_wsum<<<256, 256, 0, stream>>>(conv_w, wsum);
    gemm_wmma<<<dim3(64, 1, 10), 256, 0, stream>>>(x, conv_w + 65536, wsum,
                                                   Abuf, Bbuf);
    fuse<<<4096, 256, 0, stream>>>(x, conv_b, Abuf, Bbuf, out);
}